// GCNEncoder_31774168056018
// MI455X (gfx1250) — compile-verified
//
#include <hip/hip_runtime.h>
#include <hip/hip_bf16.h>

typedef __attribute__((ext_vector_type(2))) float v2f;
typedef __attribute__((ext_vector_type(4))) float v4f;
typedef __attribute__((ext_vector_type(8))) float v8f;

#define DIM 128
#define EPS 1e-5f
#define NBSTAT 200   // blocks for deterministic BN partial reduction

// ---------------------------------------------------------------------------
// GEMM: H[n x 128] = Z[n x 128] @ W[128 x 128], fp32 via V_WMMA_F32_16X16X4_F32
// One wave per 16x16 output tile; 4 waves per block cover 4 M-tiles, same N-tile.
// A 16x4 layout: lane m = lane&15; lane-half selects K pair; VGPR0/1 = K, K+1.
// B 4x16 layout (mirror): lane n = lane&15; lane-half selects K pair; .x/.y = K, K+1.
// C/D 16x16: VGPR i -> row (i + 8*(lane>=16)), col = lane&15.
// ---------------------------------------------------------------------------
__global__ __launch_bounds__(128) void gemm_wmma_f32(const float* __restrict__ Z,
                                                     const float* __restrict__ W,
                                                     float* __restrict__ H, int nrows) {
  const int wave = threadIdx.x >> 5;
  const int lane = threadIdx.x & 31;
  const int mtiles = (nrows + 15) >> 4;
  const int mtile = blockIdx.x * 4 + wave;
  const int ntile = blockIdx.y;
  if (mtile >= mtiles) return;                 // wave-uniform exit (EXEC stays full)

  int mrow = mtile * 16 + (lane & 15);
  if (mrow >= nrows) mrow = nrows - 1;         // clamp reads for ragged last tile
  const int khalf = (lane >> 4) * 2;           // lane-half K sub-offset
  const int ncol  = ntile * 16 + (lane & 15);  // B column for this lane
  const float* __restrict__ zrow = Z + (size_t)mrow * DIM + khalf;
  const float* __restrict__ wcol = W + (size_t)khalf * DIM + ncol;

  v8f acc = {};
#pragma unroll 8
  for (int k0 = 0; k0 < DIM; k0 += 4) {
    v2f a = *(const v2f*)(zrow + k0);          // aligned float2 (khalf, k0 even)
    v2f b;
    b.x = wcol[(size_t)k0 * DIM];
    b.y = wcol[(size_t)k0 * DIM + DIM];
    acc = __builtin_amdgcn_wmma_f32_16x16x4_f32(false, a, false, b,
                                                (short)0, acc, false, false);
  }

  const int mbase = mtile * 16 + ((lane >> 4) << 3);
  const int n = ntile * 16 + (lane & 15);
  float* __restrict__ hp = H + (size_t)mbase * DIM + n;
  if (mtile * 16 + 16 <= nrows) {
    // Full tile (always the case for N % 16 == 0): unconditional, clause-able stores.
#pragma unroll
    for (int i = 0; i < 8; ++i) hp[(size_t)i * DIM] = acc[i];
  } else {
#pragma unroll
    for (int i = 0; i < 8; ++i) {
      if (mbase + i < nrows) hp[(size_t)i * DIM] = acc[i];
    }
  }
}

// ---------------------------------------------------------------------------
// Degree / normalization helpers (computed once, reused for all 4 layers)
// ---------------------------------------------------------------------------
__global__ void fill_ones(float* __restrict__ d, int n) {
  int i = blockIdx.x * blockDim.x + threadIdx.x;
  if (i < n) d[i] = 1.0f;                      // self-loop contributes 1
}

__global__ void deg_count(const int* __restrict__ dst, int e, float* __restrict__ deg) {
  int i = blockIdx.x * blockDim.x + threadIdx.x;
  if (i < e) atomicAdd(&deg[dst[i]], 1.0f);    // integer-valued f32 adds are exact
}

__global__ void to_rsqrt(float* __restrict__ d, int n) {
  int i = blockIdx.x * blockDim.x + threadIdx.x;
  if (i < n) d[i] = rsqrtf(d[i]);              // dis = deg^{-1/2}
}

// out[r][c] = h[r][c] * dis[r]^2 + b[c]   (self-loop term + bias, also zero-inits out)
__global__ void self_bias(const float* __restrict__ H, const float* __restrict__ dis,
                          const float* __restrict__ b, float* __restrict__ Zo, int total) {
  int i = blockIdx.x * blockDim.x + threadIdx.x;
  if (i < total) {
    int r = i >> 7, c = i & (DIM - 1);
    float s = dis[r];
    Zo[i] = H[i] * (s * s) + b[c];
  }
}

// One wave (32 lanes) per edge: lane handles one float4 of the 128-float row.
__global__ __launch_bounds__(256) void scatter_edges(const float* __restrict__ H,
                                                     const float* __restrict__ dis,
                                                     const int* __restrict__ src,
                                                     const int* __restrict__ dst,
                                                     float* __restrict__ Zo, int e) {
  int t = blockIdx.x * blockDim.x + threadIdx.x;
  int ei = t >> 5, lane = t & 31;
  if (ei >= e) return;
  int s = src[ei], d = dst[ei];                // wave-uniform per edge
  float sc = dis[s] * dis[d];
  v4f v = *((const v4f*)(H + (size_t)s * DIM) + lane);   // 512B/edge gather, L2-resident
  float* op = Zo + (size_t)d * DIM + lane * 4;
  atomicAdd(op + 0, v.x * sc);
  atomicAdd(op + 1, v.y * sc);
  atomicAdd(op + 2, v.z * sc);
  atomicAdd(op + 3, v.w * sc);
}

// Fused in-place ReLU + deterministic per-column partial sums for BatchNorm.
__global__ __launch_bounds__(DIM) void relu_bnpartial(float* __restrict__ Z, int n, int rowsPerBlock,
                                                      float* __restrict__ psum, float* __restrict__ psq) {
  const int c = threadIdx.x;
  const int b = blockIdx.x;
  int r0 = b * rowsPerBlock;
  int r1 = r0 + rowsPerBlock; if (r1 > n) r1 = n;
  float s = 0.0f, q = 0.0f;
  for (int r = r0; r < r1; ++r) {
    size_t idx = (size_t)r * DIM + c;
    float v = Z[idx];
    v = v > 0.0f ? v : 0.0f;
    Z[idx] = v;
    s += v; q += v * v;
  }
  psum[b * DIM + c] = s;
  psq [b * DIM + c] = q;
}

// Ordered final reduction; folds gamma/beta/mu/invstd into per-column scale/shift.
__global__ __launch_bounds__(DIM) void bn_finalize(const float* __restrict__ psum,
                                                   const float* __restrict__ psq,
                                                   int nb, int n,
                                                   const float* __restrict__ gamma,
                                                   const float* __restrict__ beta,
                                                   float* __restrict__ scale,
                                                   float* __restrict__ shift) {
  const int c = threadIdx.x;
  float s = 0.0f, q = 0.0f;
  for (int b = 0; b < nb; ++b) { s += psum[b * DIM + c]; q += psq[b * DIM + c]; }
  const float invn = 1.0f / (float)n;
  const float mu = s * invn;
  const float var = q * invn - mu * mu;        // biased variance (jnp.var default)
  const float sc = rsqrtf(var + EPS) * gamma[c];
  scale[c] = sc;
  shift[c] = beta[c] - mu * sc;
}

__global__ void bn_apply(float* __restrict__ Z, const float* __restrict__ scale,
                         const float* __restrict__ shift, int total) {
  int i = blockIdx.x * blockDim.x + threadIdx.x;
  if (i < total) {
    int c = i & (DIM - 1);
    Z[i] = Z[i] * scale[c] + shift[c];
  }
}

// ---------------------------------------------------------------------------
extern "C" void kernel_launch(void* const* d_in, const int* in_sizes, int n_in,
                              void* d_out, int out_size, void* d_ws, size_t ws_size,
                              hipStream_t stream) {
  const float* x     = (const float*)d_in[0];
  const int*   eidx  = (const int*)  d_in[1];
  const float* Ws    = (const float*)d_in[2];
  const float* bs    = (const float*)d_in[3];
  const float* gs    = (const float*)d_in[4];
  const float* betas = (const float*)d_in[5];

  const int n = in_sizes[0] / DIM;       // 50000 nodes
  const int e = in_sizes[1] / 2;         // 800000 edges
  const int* src = eidx;
  const int* dst = eidx + e;

  float* out = (float*)d_out;
  float* ws  = (float*)d_ws;

  // Workspace layout (floats): H [n*DIM] | dis [n] | psum [NB*DIM] | psq [NB*DIM] | scale | shift
  float* H     = ws;
  float* dis   = H + (size_t)n * DIM;
  float* psum  = dis + n;
  float* psq   = psum + NBSTAT * DIM;
  float* scale = psq + NBSTAT * DIM;
  float* shift = scale + DIM;

  const int total = n * DIM;
  const int rowsPerBlock = (n + NBSTAT - 1) / NBSTAT;

  // Degrees + D^{-1/2} once (graph is fixed across layers).
  fill_ones<<<(n + 255) / 256, 256, 0, stream>>>(dis, n);
  deg_count<<<(e + 255) / 256, 256, 0, stream>>>(dst, e, dis);
  to_rsqrt <<<(n + 255) / 256, 256, 0, stream>>>(dis, n);

  const int mtiles = (n + 15) / 16;
  dim3 ggrid((mtiles + 3) / 4, DIM / 16);
  const int scatterBlocks = (int)(((long long)e * 32 + 255) / 256);

  for (int l = 0; l < 4; ++l) {
    const float* zin = (l == 0) ? x : out;
    gemm_wmma_f32<<<ggrid, 128, 0, stream>>>(zin, Ws + (size_t)l * DIM * DIM, H, n);
    self_bias    <<<(total + 255) / 256, 256, 0, stream>>>(H, dis, bs + l * DIM, out, total);
    scatter_edges<<<scatterBlocks, 256, 0, stream>>>(H, dis, src, dst, out, e);
    relu_bnpartial<<<NBSTAT, DIM, 0, stream>>>(out, n, rowsPerBlock, psum, psq);
    bn_finalize  <<<1, DIM, 0, stream>>>(psum, psq, NBSTAT, n, gs + l * DIM, betas + l * DIM, scale, shift);
    bn_apply     <<<(total + 255) / 256, 256, 0, stream>>>(out, scale, shift, total);
  }
}